// SNNModel_63745904608015
// MI455X (gfx1250) — compile-verified
//
#include <hip/hip_runtime.h>

// ---------------------------------------------------------------------------
// SNN: 199 sequential LIF steps. Strategy:
//  1) Pack 11 delayed-weight matrices to bf16 + u8 delays (L2-resident, 33MB).
//  2) Hoist input-layer currents to a fp32 WMMA GEMM over all 199 timesteps.
//  3) 5 small stream-ordered stage kernels per timestep; spike history kept
//     as per-neuron bitmasks; delayed matvec = bf16 FMA gated by (h>>D)&1.
// ---------------------------------------------------------------------------

#define NN     1024
#define TSTEPS 199
#define TP     208          // padded time columns (13 * 16) for WMMA tiling

typedef __attribute__((ext_vector_type(2))) float v2f;
typedef __attribute__((ext_vector_type(8))) float v8f;

__device__ __forceinline__ float bf2f(unsigned short u) {
  return __uint_as_float(((unsigned)u) << 16);
}

// ---------------- init: zero membrane potentials + history masks ------------
__global__ void init_state_kernel(unsigned int* p, int n) {
  int i = blockIdx.x * 256 + threadIdx.x;
  if (i < n) p[i] = 0u;
}

// ---------------- pack fp32 W -> bf16 (RNE), int32 D -> u8 ------------------
__global__ void pack_kernel(const float* __restrict__ W, const int* __restrict__ D,
                            unsigned short* __restrict__ Wp, unsigned char* __restrict__ Dp,
                            int n) {
  int i = blockIdx.x * 256 + threadIdx.x;
  if (i < n) {
    unsigned u = __float_as_uint(W[i]);
    u += 0x7FFFu + ((u >> 16) & 1u);          // round-to-nearest-even
    Wp[i] = (unsigned short)(u >> 16);
    Dp[i] = (unsigned char)D[i];
  }
}

// ---------------- build SA / RA drive matrices [NN x TP] --------------------
__global__ void xprep_kernel(const float* __restrict__ stim,
                             float* __restrict__ Xsa, float* __restrict__ Xra) {
  int idx = blockIdx.x * 256 + threadIdx.x;
  if (idx >= NN * TP) return;
  int q = idx / TP, j = idx % TP;
  float xs = 0.f, xr = 0.f;
  if (j < TSTEPS) {
    float a = stim[q * 200 + j + 1];
    float b = stim[q * 200 + j];
    xs = a;                                    // SA drive = raw stim (t=1..199)
    xr = fabsf(a - b) * 40.f;                  // RA drive = |d stim| * 40
  }
  Xsa[idx] = xs;
  Xra[idx] = xr;
}

// ---------------- fp32 WMMA GEMM: O[NNxTP] = W[NNxNN] @ X[NNxTP] ------------
// One wave per 16x16 output tile; K-loop in steps of 4 via V_WMMA_F32_16X16X4_F32.
__global__ __launch_bounds__(32) void gemm_wmma_kernel(const float* __restrict__ W,
                                                       const float* __restrict__ X,
                                                       float* __restrict__ O) {
  int m0 = blockIdx.x * 16;
  int t0 = blockIdx.y * 16;
  int lane = threadIdx.x;
  int half = lane >> 4;          // 0: K pair {0,1}, 1: K pair {2,3}
  int n    = lane & 15;          // M for A fragment, N for B/C fragments
  v8f c = {};
  const float* Wr = W + (size_t)(m0 + n) * NN;
  for (int k = 0; k < NN; k += 4) {
    v2f a = *(const v2f*)(Wr + k + 2 * half);                 // A[M=n][K=k+2h+{0,1}]
    v2f b;
    b.x = X[(size_t)(k + 2 * half)     * TP + t0 + n];        // B[K][N=n]
    b.y = X[(size_t)(k + 2 * half + 1) * TP + t0 + n];
    c = __builtin_amdgcn_wmma_f32_16x16x4_f32(false, a, false, b, (short)0, c,
                                              false, false);
  }
#pragma unroll
  for (int r = 0; r < 8; ++r)
    O[(size_t)(m0 + r + 8 * half) * TP + t0 + n] = c[r];
}

// ---------------- stage A: input-layer LIF for sa0 / ra0 --------------------
__global__ void stage0_kernel(const float* __restrict__ Isa, const float* __restrict__ Ira,
                              float* __restrict__ v, unsigned int* __restrict__ h,
                              float* __restrict__ out, int t) {
  int idx = blockIdx.x * 256 + threadIdx.x;    // 0..2047
  int pop = idx >> 10;                          // 0 = sa0, 1 = ra0
  int n   = idx & (NN - 1);
  float I = (pop ? Ira : Isa)[(size_t)n * TP + t];
  int vslot = pop ? 3 : 0;                      // v/h index (sa0=0, ra0=3)
  float* vp = v + (size_t)vslot * NN + n;
  unsigned int* hp = h + (size_t)vslot * NN + n;
  float vv = *vp * 0.9f + I;
  float s = (vv >= 1.0f) ? 1.0f : 0.0f;
  *vp = vv * (1.0f - s);
  *hp = (*hp << 1) | (unsigned)s;
  out[((size_t)vslot * NN + n) * TSTEPS + t] = s;
}

// ---------------- generic delayed-matvec LIF stage --------------------------
struct StageParams {
  const unsigned short* W[3];       // bf16 weights, row-major [NN x NN]
  const unsigned char*  D[3];       // u8 delays
  const unsigned int*   hsrc[3];    // source spike-history bitmasks [NN]
  float scale[3];
  int   nmat;
  float*        v;                  // membrane [NN]
  unsigned int* hdst;               // history to push into (nullptr for cn1)
  float*        out;                // output base for this population
};
struct StagePair { StageParams p[2]; };

// Block = 256 threads = 8 waves; one wave per output row. Histories staged in LDS.
__global__ __launch_bounds__(256) void stage_kernel(StagePair P, int t) {
  const StageParams& sp = P.p[blockIdx.y];
  __shared__ unsigned int sh[3][NN];
  for (int i = 0; i < sp.nmat; ++i)
    for (int idx = threadIdx.x; idx < NN; idx += 256)
      sh[i][idx] = sp.hsrc[i][idx];
  __syncthreads();

  int wave = threadIdx.x >> 5;
  int lane = threadIdx.x & 31;
  int row  = blockIdx.x * 8 + wave;

  float acc = 0.f;
  for (int i = 0; i < sp.nmat; ++i) {
    const unsigned short* Wr = sp.W[i] + (size_t)row * NN;
    const unsigned char*  Dr = sp.D[i] + (size_t)row * NN;
    float partial = 0.f;
#pragma unroll
    for (int jj = 0; jj < 8; ++jj) {
      int base = jj * 128 + lane * 4;                 // 4 contiguous elems/lane
      ushort4 w4 = *(const ushort4*)(Wr + base);      // global_load_b64
      uchar4  d4 = *(const uchar4*)(Dr + base);       // global_load_b32
      uint4   h4 = *(const uint4*)(&sh[i][base]);     // ds_load_b128
      if ((h4.x >> d4.x) & 1u) partial += bf2f(w4.x);
      if ((h4.y >> d4.y) & 1u) partial += bf2f(w4.y);
      if ((h4.z >> d4.z) & 1u) partial += bf2f(w4.z);
      if ((h4.w >> d4.w) & 1u) partial += bf2f(w4.w);
    }
    acc += sp.scale[i] * partial;
  }
#pragma unroll
  for (int off = 16; off > 0; off >>= 1)
    acc += __shfl_xor(acc, off, 32);

  if (lane == 0) {
    float vv = sp.v[row] * 0.9f + acc;
    float s = (vv >= 1.0f) ? 1.0f : 0.0f;
    sp.v[row] = vv * (1.0f - s);
    if (sp.hdst) sp.hdst[row] = (sp.hdst[row] << 1) | (unsigned)s;
    sp.out[(size_t)row * TSTEPS + t] = s;
  }
}

// ---------------------------------------------------------------------------
extern "C" void kernel_launch(void* const* d_in, const int* in_sizes, int n_in,
                              void* d_out, int out_size, void* d_ws, size_t ws_size,
                              hipStream_t stream) {
  const float* stim = (const float*)d_in[0];
  const float* W_SA = (const float*)d_in[1];
  const int*   D_SA = (const int*)  d_in[2];
  const float* W_RA = (const float*)d_in[3];
  const int*   D_RA = (const int*)  d_in[4];
  const float* W_CN = (const float*)d_in[5];
  const int*   D_CN = (const int*)  d_in[6];
  float* out = (float*)d_out;
  char* ws = (char*)d_ws;

  const size_t M  = (size_t)NN * NN;           // 1M elems per matrix
  const size_t XB = (size_t)NN * TP * 4;       // bytes per [NN x TP] f32 matrix

  // Workspace layout (~36.5 MB total)
  size_t off = 0;
  unsigned short* Wp  = (unsigned short*)(ws + off); off += 11 * M * 2;
  unsigned char*  Dp  = (unsigned char*) (ws + off); off += 11 * M;
  float* Xsa = (float*)(ws + off); off += XB;
  float* Xra = (float*)(ws + off); off += XB;
  float* Isa = (float*)(ws + off); off += XB;
  float* Ira = (float*)(ws + off); off += XB;
  float*        vbase = (float*)       (ws + off); off += 8 * NN * 4;
  unsigned int* hbase = (unsigned int*)(ws + off); off += 7 * NN * 4;

  unsigned int* h_sa0 = hbase + 0 * NN;
  unsigned int* h_sa1 = hbase + 1 * NN;
  unsigned int* h_sa2 = hbase + 2 * NN;
  unsigned int* h_ra0 = hbase + 3 * NN;
  unsigned int* h_ra1 = hbase + 4 * NN;
  unsigned int* h_ra2 = hbase + 5 * NN;
  unsigned int* h_cn0 = hbase + 6 * NN;

  // 1) zero v + h (15 * NN words)
  init_state_kernel<<<(15 * NN + 255) / 256, 256, 0, stream>>>((unsigned int*)vbase, 15 * NN);

  // 2) build drive matrices
  xprep_kernel<<<(NN * TP + 255) / 256, 256, 0, stream>>>(stim, Xsa, Xra);

  // 3) pack the 11 delayed-weight matrices (slots: SA1-3, RA1-3, CN0-4)
  const float* wsrc[11] = {W_SA + M, W_SA + 2 * M, W_SA + 3 * M,
                           W_RA + M, W_RA + 2 * M, W_RA + 3 * M,
                           W_CN, W_CN + M, W_CN + 2 * M, W_CN + 3 * M, W_CN + 4 * M};
  const int* dsrc[11]   = {D_SA + M, D_SA + 2 * M, D_SA + 3 * M,
                           D_RA + M, D_RA + 2 * M, D_RA + 3 * M,
                           D_CN, D_CN + M, D_CN + 2 * M, D_CN + 3 * M, D_CN + 4 * M};
  for (int i = 0; i < 11; ++i)
    pack_kernel<<<(int)((M + 255) / 256), 256, 0, stream>>>(wsrc[i], dsrc[i],
                                                            Wp + i * M, Dp + i * M, (int)M);

  // 4) hoisted input-layer GEMMs over all 199 timesteps (fp32 WMMA)
  gemm_wmma_kernel<<<dim3(NN / 16, TP / 16), 32, 0, stream>>>(W_SA, Xsa, Isa);
  gemm_wmma_kernel<<<dim3(NN / 16, TP / 16), 32, 0, stream>>>(W_RA, Xra, Ira);

  // 5) build per-stage parameter blocks (constant across timesteps)
  const size_t ONP = (size_t)NN * TSTEPS;      // out stride per population
  auto slotW = [&](int i) { return (const unsigned short*)(Wp + (size_t)i * M); };
  auto slotD = [&](int i) { return (const unsigned char*) (Dp + (size_t)i * M); };

  StagePair sB = {}; // sa1, ra1
  sB.p[0] = { {slotW(0)}, {slotD(0)}, {h_sa0}, {1.f}, 1, vbase + 1 * NN, h_sa1, out + 1 * ONP };
  sB.p[1] = { {slotW(3)}, {slotD(3)}, {h_ra0}, {1.f}, 1, vbase + 4 * NN, h_ra1, out + 4 * ONP };

  StagePair sC = {}; // sa2, ra2
  sC.p[0] = { {slotW(1), slotW(2)}, {slotD(1), slotD(2)}, {h_sa0, h_sa1},
              {10.f, -3.f}, 2, vbase + 2 * NN, h_sa2, out + 2 * ONP };
  sC.p[1] = { {slotW(4), slotW(5)}, {slotD(4), slotD(5)}, {h_ra0, h_ra1},
              {10.f, -3.f}, 2, vbase + 5 * NN, h_ra2, out + 5 * ONP };

  StagePair sD = {}; // cn0
  sD.p[0] = { {slotW(6), slotW(8)}, {slotD(6), slotD(8)}, {h_sa2, h_ra2},
              {1.f, 1.f}, 2, vbase + 6 * NN, h_cn0, out + 6 * ONP };

  StagePair sE = {}; // cn1 (no history buffer)
  sE.p[0] = { {slotW(7), slotW(9), slotW(10)}, {slotD(7), slotD(9), slotD(10)},
              {h_sa2, h_ra2, h_cn0}, {5.f, 5.f, -6.f}, 3,
              vbase + 7 * NN, nullptr, out + 7 * ONP };

  // 6) sequential timestep loop: 5 dependency stages per step
  for (int t = 0; t < TSTEPS; ++t) {
    stage0_kernel<<<(2 * NN + 255) / 256, 256, 0, stream>>>(Isa, Ira, vbase, hbase, out, t);
    stage_kernel<<<dim3(NN / 8, 2), 256, 0, stream>>>(sB, t);
    stage_kernel<<<dim3(NN / 8, 2), 256, 0, stream>>>(sC, t);
    stage_kernel<<<dim3(NN / 8, 1), 256, 0, stream>>>(sD, t);
    stage_kernel<<<dim3(NN / 8, 1), 256, 0, stream>>>(sE, t);
  }
}